// Attention_5145370821223
// MI455X (gfx1250) — compile-verified
//
#include <hip/hip_runtime.h>
#include <math.h>
#include <stdint.h>

typedef float v2f __attribute__((ext_vector_type(2)));
typedef float v8f __attribute__((ext_vector_type(8)));
typedef unsigned int v4u __attribute__((ext_vector_type(4)));
typedef int v4i __attribute__((ext_vector_type(4)));
typedef int v8i __attribute__((ext_vector_type(8)));

#define BATCH 16
#define SEQ   512
#define DIMX  78
#define NH    6
#define DH    13
#define DHP   16      // DH padded to 16 (zeros) so WMMA K-padding is free
#define QKVW  (NH*3*DH)   // 234
#define OHW   80          // H*DH=78 padded to 80 for K-chunks of 4
#define BIGF  3.402823466e38f
#define KST   18          // LDS row stride for K tiles: even (b64 frag loads)
                          // and 18*lm mod 64 distinct for lm<16 -> no conflicts
#define PST   516         // LDS stride for P: even, 516%64=4 -> conflict-free
#define WPAD  (18*20*32*2)  // packed W_qkv, fragment order  (23040)
#define WOPAD (5*20*32*2)   // packed W_out, fragment order  (6400)

// D = A(16x4 f32) * B(4x16 f32) + C(16x16 f32), full fp32 on the matrix pipe.
__device__ __forceinline__ v8f wmma_f32(v2f a, v2f b, v8f c) {
  return __builtin_amdgcn_wmma_f32_16x16x4_f32(false, a, false, b, (short)0, c,
                                               false, false);
}

// ---------------------------------------------------------------------------
// TDM: 2-D tensor_load_to_lds.  tile_w (<=16) x tile_h fp32 elements, global
// row stride `stride` elements.  pad_dwords: LDS padding inserted after every
// 16 dwords stored (0 = none) -> turns a 16-wide row into LDS stride 16+pad.
// Descriptor fields per CDNA5 ISA.  Wave-uniform args.
// ---------------------------------------------------------------------------
__device__ __forceinline__ void tdm_load_2d(unsigned lds_off, const void* gptr,
                                            unsigned tile_w, unsigned tile_h,
                                            unsigned stride,
                                            unsigned pad_dwords) {
  unsigned long long ga = (unsigned long long)(uintptr_t)gptr;
  v4u g0;
  g0[0] = 1u;                                   // count=1, user descriptor
  g0[1] = lds_off;                              // lds_addr (bytes)
  g0[2] = (unsigned)(ga & 0xffffffffu);         // global_addr[31:0]
  g0[3] = (unsigned)((ga >> 32) & 0x01ffffffu)  // global_addr[56:32]
          | (2u << 30);                         // type=2 ("image")
  unsigned d0 = (2u << 16);                     // data_size = 4 bytes
  if (pad_dwords) d0 |= (1u << 20)              // pad_enable
                     |  (3u << 22)              // pad_interval: 16 dwords
                     |  ((pad_dwords - 1u) << 25);  // pad_amount code
  v8i g1;
  g1[0] = (int)d0;
  g1[1] = (int)(tile_w << 16);   // tensor_dim0[15:0] (== tile_w)
  g1[2] = (int)(tile_h << 16);   // tensor_dim0 hi=0 | tensor_dim1[15:0]
  g1[3] = (int)(tile_w << 16);   // tensor_dim1 hi=0 | tile_dim0
  g1[4] = (int)tile_h;           // tile_dim1 (tile_dim2 = 0)
  g1[5] = (int)stride;           // tensor_dim0_stride[31:0] (elements)
  g1[6] = 0;
  g1[7] = 0;
  v4i z4 = {};
#if __clang_major__ >= 23
  v8i z8 = {};
  __builtin_amdgcn_tensor_load_to_lds(g0, g1, z4, z4, z8, 0);
#else
  __builtin_amdgcn_tensor_load_to_lds(g0, g1, z4, z4, 0);
#endif
}

// ---------------------------------------------------------------------------
// Kernel 0: pack weights into WMMA-fragment-ordered, zero-padded layouts.
// Wp [t][j][lane]{2}: B-fragment of W_qkv for tile t=(h*3+comp), K-chunk j.
//   element e of lane: k = 4j + 2*(lane>>4) + e, col = lane&15 within tile.
//   Q tiles (comp==0) pre-scaled by DH^-0.5.
// Wop[ct][j][lane]{2}: same for W_out column-tile ct.   bop[80]: padded bias.
// ---------------------------------------------------------------------------
__global__ __launch_bounds__(256) void pack_kernel(
    const float* __restrict__ Wqkv, const float* __restrict__ Wout,
    const float* __restrict__ bout, float* __restrict__ Wp,
    float* __restrict__ Wop, float* __restrict__ bop) {
  int i = blockIdx.x * 256 + threadIdx.x;
  if (i < WPAD) {
    int e = i & 1, lane = (i >> 1) & 31, jt = i >> 6;
    int j = jt % 20, t = jt / 20;
    int h = t / 3, comp = t % 3;
    int k = j * 4 + 2 * (lane >> 4) + e;
    int c = lane & 15;
    float v = 0.f;
    if (k < DIMX && c < DH) v = Wqkv[k * QKVW + h * 3 * DH + comp * DH + c];
    if (comp == 0) v *= 0.2773500981126146f;   // fold DH^-0.5 into Q weights
    Wp[i] = v;
  } else if (i < WPAD + WOPAD) {
    int q = i - WPAD;
    int e = q & 1, lane = (q >> 1) & 31, jc = q >> 6;
    int j = jc % 20, ct = jc / 20;
    int k = j * 4 + 2 * (lane >> 4) + e;
    int col = ct * 16 + (lane & 15);
    Wop[q] = (k < DIMX && col < DIMX) ? Wout[k * DIMX + col] : 0.f;
  } else if (i < WPAD + WOPAD + 80) {
    int c = i - WPAD - WOPAD;
    bop[c] = (c < DIMX) ? bout[c] : 0.f;
  }
}

// ---------------------------------------------------------------------------
// Kernel 1: qkv = x @ W_qkv written as padded per-head Q,K,V (Q pre-scaled).
// grid = B*N/16 blocks, 64 threads (2 waves); wave w handles heads [3w,3w+3).
// B-fragments are contiguous v2f loads (clause-batched global_load_b64).
// ---------------------------------------------------------------------------
__global__ __launch_bounds__(64) void qkv_kernel(const float* __restrict__ x,
                                                 const float* __restrict__ Wp,
                                                 float* __restrict__ Q,
                                                 float* __restrict__ K,
                                                 float* __restrict__ V) {
  __shared__ alignas(16) float Xs[16 * 80];  // 16 rows of x, padded 78->80
  const int tid  = threadIdx.x;
  const int wave = tid >> 5;
  const int lane = tid & 31;
  const int half = lane >> 4;
  const int lm   = lane & 15;
  const int b  = blockIdx.x >> 5;
  const int n0 = (blockIdx.x & 31) << 4;

  for (int i = tid; i < 16 * 80; i += 64) {
    int r = i / 80, c = i % 80;
    Xs[i] = (c < DIMX) ? x[((long)(b * SEQ) + n0 + r) * DIMX + c] : 0.f;
  }
  __syncthreads();

  for (int h = wave * 3; h < wave * 3 + 3; ++h)
    for (int comp = 0; comp < 3; ++comp) {
      const v2f* wp = (const v2f*)Wp + ((h * 3 + comp) * 20) * 32 + lane;
      v8f acc = {};
#pragma unroll
      for (int j = 0; j < 20; ++j) {
        const int kk = j * 4 + 2 * half;
        v2f a  = *(const v2f*)&Xs[lm * 80 + kk];   // ds_load_b64
        v2f bb = wp[j * 32];                       // global_load_b64, coalesced
        acc = wmma_f32(a, bb, acc);
      }
      float* dst = (comp == 0) ? Q : (comp == 1) ? K : V;
      const long base = (long)(b * NH + h) * SEQ + n0;
#pragma unroll
      for (int r = 0; r < 8; ++r)
        dst[(base + r + 8 * half) * DHP + lm] = acc[r];
    }
}

// ---------------------------------------------------------------------------
// Kernel 2: fused masked softmax attention + adjacency blend for one
// (batch, head, 16-query tile).  256 threads = 8 waves; wave w owns columns
// [w*64, w*64+64).  K/V staged to LDS by the Tensor Data Mover.
// ---------------------------------------------------------------------------
__global__ __launch_bounds__(256) void attn_kernel(
    const float* __restrict__ Q, const float* __restrict__ K,
    const float* __restrict__ V, const unsigned char* __restrict__ mask,
    const float* __restrict__ adj, float* __restrict__ OH) {
  __shared__ alignas(16) float Kb[SEQ * KST];  // TDM-padded K; reused as P
  __shared__ alignas(16) float Vb[SEQ * DHP];
  __shared__ float red[8 * 16];
  __shared__ float rstat[32];       // [0,16) row max, [16,32) 1/rowsum
  __shared__ float Ored[16 * 16];

  const int tid  = threadIdx.x;
  const int wave = tid >> 5;
  const int lane = tid & 31;
  const int half = lane >> 4;
  const int lm   = lane & 15;
  const int qt = blockIdx.x & 31;
  const int h  = (blockIdx.x >> 5) % NH;
  const int b  = blockIdx.x / (32 * NH);
  const int q0 = qt << 4;
  const int colbase = wave * 64;

  // Async-stage K (LDS row stride 18 via TDM padding) and V (dense).
  if (wave == 0) {
    const float* Kg = K + (long)((b * NH + h) * SEQ) * DHP;
    const float* Vg = V + (long)((b * NH + h) * SEQ) * DHP;
    tdm_load_2d((unsigned)(uintptr_t)(void*)Kb, Kg, DHP, SEQ, DHP, KST - DHP);
    tdm_load_2d((unsigned)(uintptr_t)(void*)Vb, Vg, DHP, SEQ, DHP, 0);
  }
  Ored[tid] = 0.f;

  // Prefetch this block's adjacency segment (cold HBM stream consumed later).
#pragma unroll
  for (int r = 0; r < 8; ++r)
    __builtin_prefetch(
        &adj[((long)(b * SEQ) + q0 + r + 8 * half) * SEQ + colbase + lm * 4],
        0, 0);

  const bool mq = mask[b * SEQ + q0 + lm] != 0;
  const unsigned qbits = (unsigned)__ballot(mq);   // bit M = query-row mask
  bool km[4];
#pragma unroll
  for (int ct = 0; ct < 4; ++ct)
    km[ct] = mask[b * SEQ + colbase + ct * 16 + lm] != 0;

  const float* Qg = Q + (long)((b * NH + h) * SEQ + q0) * DHP;
  v2f aq[4];
#pragma unroll
  for (int j = 0; j < 4; ++j)
    aq[j] = *(const v2f*)(Qg + lm * DHP + j * 4 + 2 * half);  // b64 loads
  if (wave == 0) __builtin_amdgcn_s_wait_tensorcnt(0);
  __syncthreads();

  // ---- scores S = (Q*scale) @ K^T, 4 column tiles per wave ----
  v8f s[4];
#pragma unroll
  for (int ct = 0; ct < 4; ++ct) {
    v8f acc = {};
    const int col = colbase + ct * 16 + lm;
#pragma unroll
    for (int j = 0; j < 4; ++j) {
      v2f bb = *(const v2f*)&Kb[col * KST + j * 4 + 2 * half];  // ds_load_b64
      acc = wmma_f32(aq[j], bb, acc);
    }
    s[ct] = acc;
  }

  // ---- mask + row max (C layout: row = r + 8*half, col across 16 lanes) ----
  float rm[8];
#pragma unroll
  for (int r = 0; r < 8; ++r) rm[r] = -BIGF;
#pragma unroll
  for (int ct = 0; ct < 4; ++ct)
#pragma unroll
    for (int r = 0; r < 8; ++r) {
      const int M = r + 8 * half;
      const bool m = ((qbits >> M) & 1) && km[ct];
      const float v = m ? s[ct][r] : -BIGF;
      s[ct][r] = v;
      rm[r] = fmaxf(rm[r], v);
    }
#pragma unroll
  for (int off = 8; off; off >>= 1)
#pragma unroll
    for (int r = 0; r < 8; ++r)
      rm[r] = fmaxf(rm[r], __shfl_xor(rm[r], off, 16));
  if (lm == 0)
#pragma unroll
    for (int r = 0; r < 8; ++r) red[wave * 16 + r + 8 * half] = rm[r];
  __syncthreads();
  if (tid < 16) {
    float mx = -BIGF;
#pragma unroll
    for (int w = 0; w < 8; ++w) mx = fmaxf(mx, red[w * 16 + tid]);
    rstat[tid] = mx;
  }
  __syncthreads();

  // ---- exp + row sum ----
  float rowmax[8], rs[8];
#pragma unroll
  for (int r = 0; r < 8; ++r) { rowmax[r] = rstat[r + 8 * half]; rs[r] = 0.f; }
#pragma unroll
  for (int ct = 0; ct < 4; ++ct)
#pragma unroll
    for (int r = 0; r < 8; ++r) {
      const float e = __expf(s[ct][r] - rowmax[r]);
      s[ct][r] = e;
      rs[r] += e;
    }
#pragma unroll
  for (int off = 8; off; off >>= 1)
#pragma unroll
    for (int r = 0; r < 8; ++r) rs[r] += __shfl_xor(rs[r], off, 16);
  if (lm == 0)
#pragma unroll
    for (int r = 0; r < 8; ++r) red[wave * 16 + r + 8 * half] = rs[r];
  __syncthreads();
  if (tid < 16) {
    float sum = 0.f;
#pragma unroll
    for (int w = 0; w < 8; ++w) sum += red[w * 16 + tid];
    rstat[16 + tid] = 1.f / sum;
  }
  __syncthreads();
  float rinv[8];
#pragma unroll
  for (int r = 0; r < 8; ++r) rinv[r] = rstat[16 + r + 8 * half];

  // ---- normalize, blend 0.5*masked(adj), stage P into LDS (overlays Kb) ----
  float* P = Kb;
#pragma unroll
  for (int ct = 0; ct < 4; ++ct) {
    const int col = colbase + ct * 16 + lm;
#pragma unroll
    for (int r = 0; r < 8; ++r) {
      const int M = r + 8 * half;
      const bool m = ((qbits >> M) & 1) && km[ct];
      const float a_ = m ? adj[((long)(b * SEQ) + q0 + M) * SEQ + col] : 0.f;
      P[M * PST + col] = s[ct][r] * rinv[r] + 0.5f * a_;
    }
  }
  __syncthreads();

  // ---- O += P @ V over this wave's K-range; LDS-atomic cross-wave reduce ----
  v8f o = {};
#pragma unroll
  for (int c = 0; c < 16; ++c) {
    const int k0 = wave * 64 + c * 4 + 2 * half;
    v2f a = *(const v2f*)&P[lm * PST + k0];          // ds_load_b64
    v2f bb; bb.x = Vb[k0 * DHP + lm]; bb.y = Vb[(k0 + 1) * DHP + lm];
    o = wmma_f32(a, bb, o);
  }
#pragma unroll
  for (int r = 0; r < 8; ++r)
    atomicAdd(&Ored[(r + 8 * half) * 16 + lm], o[r]);
  __syncthreads();

  if (tid < 16 * DH) {
    const int row = tid / DH, c = tid % DH;
    OH[((long)(b * SEQ) + q0 + row) * OHW + h * DH + c] = Ored[row * 16 + c];
  }
  if (h == 0 && tid >= 224) {            // zero OH pad columns 78,79 once
    const int i = tid - 224, row = i >> 1, c = i & 1;
    OH[((long)(b * SEQ) + q0 + row) * OHW + DIMX + c] = 0.f;
  }
}

// ---------------------------------------------------------------------------
// Kernel 3: out = OH @ W_out + b_out.  One wave per 16x16 output tile.
// ---------------------------------------------------------------------------
__global__ __launch_bounds__(32) void proj_kernel(const float* __restrict__ OH,
                                                  const float* __restrict__ Wop,
                                                  const float* __restrict__ bop,
                                                  float* __restrict__ out) {
  const int lane = threadIdx.x;
  const int half = lane >> 4;
  const int lm   = lane & 15;
  const int ct = blockIdx.x % 5;
  const long R0 = (long)(blockIdx.x / 5) * 16;
  const int col = ct * 16 + lm;
  const v2f* wop = (const v2f*)Wop + (ct * 20) * 32 + lane;
  v8f acc = {};
#pragma unroll
  for (int j = 0; j < 20; ++j) {
    v2f a  = *(const v2f*)(OH + (R0 + lm) * OHW + j * 4 + 2 * half);
    v2f bb = wop[j * 32];                          // global_load_b64, coalesced
    acc = wmma_f32(a, bb, acc);
  }
  if (col < DIMX) {
    const float bo = bop[col];
#pragma unroll
    for (int r = 0; r < 8; ++r)
      out[(R0 + r + 8 * half) * DIMX + col] = acc[r] + bo;
  }
}

extern "C" void kernel_launch(void* const* d_in, const int* in_sizes, int n_in,
                              void* d_out, int out_size, void* d_ws,
                              size_t ws_size, hipStream_t stream) {
  (void)in_sizes; (void)n_in; (void)out_size; (void)ws_size;
  const float*         x    = (const float*)d_in[0];
  const unsigned char* mask = (const unsigned char*)d_in[1];  // numpy bool = 1B
  const float*         adj  = (const float*)d_in[2];
  const float*         Wqkv = (const float*)d_in[3];
  const float*         Wout = (const float*)d_in[4];
  const float*         bout = (const float*)d_in[5];

  float* Q = (float*)d_ws;
  const size_t per = (size_t)BATCH * NH * SEQ * DHP;   // 786432 floats each
  float* K   = Q + per;
  float* V   = K + per;
  float* OH  = V + per;                                // BATCH*SEQ*80 floats
  float* Wp  = OH + (size_t)BATCH * SEQ * OHW;
  float* Wop = Wp + WPAD;
  float* bop = Wop + WOPAD;

  pack_kernel<<<(WPAD + WOPAD + 80 + 255) / 256, 256, 0, stream>>>(
      Wqkv, Wout, bout, Wp, Wop, bop);
  qkv_kernel<<<BATCH * SEQ / 16, 64, 0, stream>>>(x, Wp, Q, K, V);
  attn_kernel<<<BATCH * NH * (SEQ / 16), 256, 0, stream>>>(Q, K, V, mask, adj,
                                                           OH);
  proj_kernel<<<(BATCH * SEQ / 16) * 5, 32, 0, stream>>>(OH, Wop, bop,
                                                         (float*)d_out);
}